// EmbeddingShard_7069516169405
// MI455X (gfx1250) — compile-verified
//
#include <hip/hip_runtime.h>

// ---------------------------------------------------------------------------
// Embedding gather: out[n,:] = (W_full[tok[n],:] + sum_s b[s,:]) / SHARDS
// Bandwidth-bound (≈1.07 GB traffic, ~46 µs floor @ 23.3 TB/s).
// CDNA5 path: GLOBAL_LOAD_ASYNC_TO_LDS_B128 double-buffered DMA gather,
// s_wait_asynccnt pipelining, non-temporal b128 stores.
// ---------------------------------------------------------------------------

#define OUT_DIM   4096
#define SHARDS    8
#define TPB       256                      // 8 waves of 32
#define VEC       4                        // floats per lane per round (b128)
#define ROUNDS    (OUT_DIM / (TPB * VEC))  // 4 rounds cover one 16 KB row
#define ROW_BYTES (OUT_DIM * 4)            // 16384

typedef __attribute__((ext_vector_type(4))) float v4f;

// --- Kernel 1: bscaled[j] = 0.125f * sum_s b[s][j]  (16 KB, lives in d_ws) ---
__global__ void bias_prescale_kernel(const float* __restrict__ b,
                                     float* __restrict__ bscaled) {
    int j = blockIdx.x * blockDim.x + threadIdx.x;
    if (j < OUT_DIM) {
        float s = 0.0f;
#pragma unroll
        for (int k = 0; k < SHARDS; ++k) s += b[k * OUT_DIM + j];
        bscaled[j] = s * 0.125f;
    }
}

// --- Kernel 2: double-buffered async-DMA gather ------------------------------
__global__ __launch_bounds__(TPB) void embed_gather_kernel(
    const int* __restrict__ tok, const float* __restrict__ W,
    const float* __restrict__ bscaled, float* __restrict__ out, int n) {

    __shared__ __align__(16) float buf[2][OUT_DIM];   // 32 KB double buffer

    const int tid = threadIdx.x;

    // Pre-scaled bias cached in registers: same columns for every token.
    v4f bias[ROUNDS];
#pragma unroll
    for (int r = 0; r < ROUNDS; ++r)
        bias[r] = *(const v4f*)(bscaled + r * (TPB * VEC) + tid * VEC);

    // Per-lane LDS byte offsets (generic-pointer low bits == LDS offset).
    const unsigned lds0 = (unsigned)(size_t)(void*)&buf[0][tid * VEC];
    const unsigned lds1 = (unsigned)(size_t)(void*)&buf[1][tid * VEC];

    long long t = blockIdx.x;
    const long long stride = gridDim.x;
    if (t >= n) return;

    // Prologue: DMA first row into buffer 0 (4 async b128 per wave -> ASYNCcnt=4).
    {
        long long id0 = tok[t];
        const char* src0 = (const char*)W + id0 * (long long)ROW_BYTES
                         + (size_t)tid * 16u;
        asm volatile("global_load_async_to_lds_b128 %0, %1, off"
                     :: "v"(lds0), "v"(src0) : "memory");
        asm volatile("global_load_async_to_lds_b128 %0, %1, off offset:4096"
                     :: "v"(lds0), "v"(src0) : "memory");
        asm volatile("global_load_async_to_lds_b128 %0, %1, off offset:8192"
                     :: "v"(lds0), "v"(src0) : "memory");
        asm volatile("global_load_async_to_lds_b128 %0, %1, off offset:12288"
                     :: "v"(lds0), "v"(src0) : "memory");
    }

    int cur = 0;
    for (; t < n; t += stride) {
        const long long tn = t + stride;
        if (tn < n) {
            // Refill the other buffer. WAR fence: our ds_loads from it (previous
            // iteration) must have completed before the DMA overwrites it.
            long long idn = tok[tn];
            const char* srcn = (const char*)W + idn * (long long)ROW_BYTES
                             + (size_t)tid * 16u;
            const unsigned dstn = cur ? lds0 : lds1;
            asm volatile("s_wait_dscnt 0" ::: "memory");
            asm volatile("global_load_async_to_lds_b128 %0, %1, off"
                         :: "v"(dstn), "v"(srcn) : "memory");
            asm volatile("global_load_async_to_lds_b128 %0, %1, off offset:4096"
                         :: "v"(dstn), "v"(srcn) : "memory");
            asm volatile("global_load_async_to_lds_b128 %0, %1, off offset:8192"
                         :: "v"(dstn), "v"(srcn) : "memory");
            asm volatile("global_load_async_to_lds_b128 %0, %1, off offset:12288"
                         :: "v"(dstn), "v"(srcn) : "memory");
            // Retire the older 4 DMAs -> current buffer is ready.
            asm volatile("s_wait_asynccnt 4" ::: "memory");
        } else {
            asm volatile("s_wait_asynccnt 0" ::: "memory");
        }

        // Consume current buffer: LDS b128 load, fused scale+bias, NT b128 store.
        const float* p = &buf[cur][tid * VEC];
        float* o = out + (size_t)t * OUT_DIM + (size_t)tid * VEC;
#pragma unroll
        for (int r = 0; r < ROUNDS; ++r) {
            v4f w = *(const v4f*)(p + r * (TPB * VEC));
            v4f res = w * 0.125f + bias[r];
            __builtin_nontemporal_store(res, (v4f*)(o + r * (TPB * VEC)));
        }
        cur ^= 1;
    }
}

// ---------------------------------------------------------------------------
extern "C" void kernel_launch(void* const* d_in, const int* in_sizes, int n_in,
                              void* d_out, int out_size, void* d_ws, size_t ws_size,
                              hipStream_t stream) {
    (void)n_in; (void)out_size; (void)ws_size;
    const int*   tok = (const int*)d_in[0];    // x: [16,2048] int32 token ids
    const float* W   = (const float*)d_in[1];  // [8, 6300, 4096] fp32
    const float* b   = (const float*)d_in[2];  // [8, 4096] fp32
    float* out       = (float*)d_out;          // [n_tok, 4096] fp32
    float* bscaled   = (float*)d_ws;           // 4096 floats scratch

    const int n_tok = in_sizes[0];             // 32768

    bias_prescale_kernel<<<(OUT_DIM + 255) / 256, 256, 0, stream>>>(b, bscaled);

    int blocks = n_tok < 4096 ? n_tok : 4096;  // ~8 tokens/block, deep pipeline
    embed_gather_kernel<<<blocks, TPB, 0, stream>>>(tok, W, bscaled, out, n_tok);
}